// TensorNetCore_68058051772518
// MI455X (gfx1250) — compile-verified
//
#include <hip/hip_runtime.h>
#include <math.h>

// ---------------- problem constants ----------------
#define NN 10000
#define EE 160000
#define FF 64
#define NF (NN * FF)          // 640000 per-channel elements
#define PI_F 3.14159265358979323846f

// LDS row strides padded to kill bank conflicts on the 64-bank LDS:
// bank = (stride*row + k) % 64 ; stride ≡ 4 (mod 64) makes all 32 lanes
// (row = lane&15, k-offset = 2*(lane>>4)) land on distinct banks, and the
// even stride keeps paired loads 8-byte aligned (ds_load_b64).
#define H1S 68     // 64 + 4
#define H2S 132    // 128 + 4
#define RDS 196    // 192 + 4

typedef float v2f __attribute__((ext_vector_type(2)));
typedef float v8f __attribute__((ext_vector_type(8)));

// fp32 WMMA: D(16x16,f32) = A(16x4,f32) x B(4x16,f32) + C
__device__ __forceinline__ v8f wmma4(v2f a, v2f b, v8f c) {
    return __builtin_amdgcn_wmma_f32_16x16x4_f32(
        /*neg_a=*/false, a, /*neg_b=*/false, b,
        /*c_mod=*/(short)0, c, /*reuse_a=*/false, /*reuse_b=*/false);
}

__device__ __forceinline__ float silu(float x) {
    return x / (1.0f + expf(-x));
}

// compact channels: [d, a01,a02,a12, s00,s11,s22, s01,s02,s12]
__device__ __forceinline__ void recon3(const float c[10], float t[9]) {
    float d = c[0];
    t[0] = d + c[4];  t[4] = d + c[5];  t[8] = d + c[6];
    t[1] = c[1] + c[7];  t[3] = c[7] - c[1];
    t[2] = c[2] + c[8];  t[6] = c[8] - c[2];
    t[5] = c[3] + c[9];  t[7] = c[9] - c[3];
}

__device__ __forceinline__ void mm3(const float a[9], const float b[9], float c[9]) {
#pragma unroll
    for (int i = 0; i < 3; ++i)
#pragma unroll
        for (int j = 0; j < 3; ++j) {
            float s = 0.f;
#pragma unroll
            for (int k = 0; k < 3; ++k) s += a[i * 3 + k] * b[k * 3 + j];
            c[i * 3 + j] = s;
        }
}

// ---------------- K1: normalize + compact decomposition ----------------
__global__ void prep_kernel(const float* __restrict__ X,
                            float* __restrict__ Xn,
                            float* __restrict__ P) {
    int idx = blockIdx.x * blockDim.x + threadIdx.x;
    if (idx >= NF) return;
    float x[9]; float nrm = 0.f;
#pragma unroll
    for (int i = 0; i < 9; ++i) { x[i] = X[(size_t)idx * 9 + i]; nrm += x[i] * x[i]; }
    float inv = 1.0f / (nrm + 1.0f);
#pragma unroll
    for (int i = 0; i < 9; ++i) { x[i] *= inv; Xn[(size_t)idx * 9 + i] = x[i]; }
    float dm = (x[0] + x[4] + x[8]) * (1.0f / 3.0f);
    P[0 * NF + idx] = dm;
    P[1 * NF + idx] = 0.5f * (x[1] - x[3]);
    P[2 * NF + idx] = 0.5f * (x[2] - x[6]);
    P[3 * NF + idx] = 0.5f * (x[5] - x[7]);
    P[4 * NF + idx] = x[0] - dm;
    P[5 * NF + idx] = x[4] - dm;
    P[6 * NF + idx] = x[8] - dm;
    P[7 * NF + idx] = 0.5f * (x[1] + x[3]);
    P[8 * NF + idx] = 0.5f * (x[2] + x[6]);
    P[9 * NF + idx] = 0.5f * (x[5] + x[7]);
}

// ---------------- K2/K5: per-channel feature mixing GEMM (WMMA) ------------
// Q[ch][n][g] = sum_f P[ch][n][f] * Lsel[f][g],  Lsel: ch0->La, ch1-3->Lb, ch4-9->Lc
__global__ __launch_bounds__(128) void mix_kernel(const float* __restrict__ P,
                                                  float* __restrict__ Q,
                                                  const float* __restrict__ La,
                                                  const float* __restrict__ Lb,
                                                  const float* __restrict__ Lc) {
    const int lane = threadIdx.x & 31;
    const int wave = threadIdx.x >> 5;
    const int ch = blockIdx.y;
    const int r0 = blockIdx.x * 16;
    const float* W = (ch == 0) ? La : (ch < 4 ? Lb : Lc);
    const float* A = P + (size_t)ch * NF;
    float* D = Q + (size_t)ch * NF;
    const int row = lane & 15;
    const int hi = lane >> 4;
    const int kb = hi * 2;
    const int col = wave * 16 + (lane & 15);
    v8f acc = {};
#pragma unroll
    for (int k = 0; k < FF; k += 4) {
        v2f a, b;
        a.x = A[(size_t)(r0 + row) * FF + k + kb];
        a.y = A[(size_t)(r0 + row) * FF + k + kb + 1];
        b.x = W[(k + kb) * FF + col];
        b.y = W[(k + kb + 1) * FF + col];
        acc = wmma4(a, b, acc);
    }
#pragma unroll
    for (int r = 0; r < 8; ++r)
        D[(size_t)(r0 + r + 8 * hi) * FF + col] = acc[r];
}

// ---------------- K3: fused edge MLP (WMMA) + compact message scatter ------
__global__ __launch_bounds__(32) void edge_kernel(
    const float* __restrict__ rbf, const float* __restrict__ dij,
    const int* __restrict__ pi,
    const float* __restrict__ W1, const float* __restrict__ b1,
    const float* __restrict__ W2, const float* __restrict__ b2,
    const float* __restrict__ W3, const float* __restrict__ b3,
    const float* __restrict__ Q, float* __restrict__ M) {
    __shared__ float h1[16 * H1S];    // layer-1 activations (A for layer 2)
    __shared__ float h2[16 * H2S];    // layer-2 activations (A for layer 3)
    __shared__ float rad[16 * RDS];   // final per-edge radial scales
    __shared__ float cc[16];          // cosine cutoff per edge

    const int lane = threadIdx.x;
    const int e0 = blockIdx.x * 16;
    const int row = lane & 15;
    const int hi = lane >> 4;
    const int kb = hi * 2;
    const int col16 = lane & 15;

    if (lane < 16) {
        float d = dij[e0 + lane];
        cc[lane] = 0.5f * (cosf(d * (PI_F / 0.5f)) + 1.0f) * (d < 0.5f ? 1.0f : 0.0f);
    }

    // ---- layer 1: [16,32] @ W1[32,64] -> silu -> h1 ----
#pragma unroll
    for (int j = 0; j < 4; ++j) {
        const int col = j * 16 + col16;
        v8f acc = {};
#pragma unroll
        for (int k = 0; k < 32; k += 4) {
            v2f a, b;
            a.x = rbf[(size_t)(e0 + row) * 32 + k + kb];
            a.y = rbf[(size_t)(e0 + row) * 32 + k + kb + 1];
            b.x = W1[(k + kb) * 64 + col];
            b.y = W1[(k + kb + 1) * 64 + col];
            acc = wmma4(a, b, acc);
        }
        float bias = b1[col];
#pragma unroll
        for (int r = 0; r < 8; ++r)
            h1[(r + 8 * hi) * H1S + col] = silu(acc[r] + bias);
    }
    __syncthreads();

    // ---- layer 2: h1[16,64] @ W2[64,128] -> silu -> h2 ----
#pragma unroll
    for (int j = 0; j < 8; ++j) {
        const int col = j * 16 + col16;
        v8f acc = {};
#pragma unroll
        for (int k = 0; k < 64; k += 4) {
            v2f a, b;
            a.x = h1[row * H1S + k + kb];
            a.y = h1[row * H1S + k + kb + 1];
            b.x = W2[(k + kb) * 128 + col];
            b.y = W2[(k + kb + 1) * 128 + col];
            acc = wmma4(a, b, acc);
        }
        float bias = b2[col];
#pragma unroll
        for (int r = 0; r < 8; ++r)
            h2[(r + 8 * hi) * H2S + col] = silu(acc[r] + bias);
    }
    __syncthreads();

    // ---- layer 3: h2[16,128] @ W3[128,192] -> silu * C -> rad ----
#pragma unroll
    for (int j = 0; j < 12; ++j) {
        const int col = j * 16 + col16;
        v8f acc = {};
#pragma unroll
        for (int k = 0; k < 128; k += 4) {
            v2f a, b;
            a.x = h2[row * H2S + k + kb];
            a.y = h2[row * H2S + k + kb + 1];
            b.x = W3[(k + kb) * 192 + col];
            b.y = W3[(k + kb + 1) * 192 + col];
            acc = wmma4(a, b, acc);
        }
        float bias = b3[col];
#pragma unroll
        for (int r = 0; r < 8; ++r) {
            int er = r + 8 * hi;
            rad[er * RDS + col] = silu(acc[r] + bias) * cc[er];
        }
    }
    __syncthreads();

    // ---- compact message pass: gather Q[*, dst, f], scatter-add M[*, src, f]
    for (int e = 0; e < 16; ++e) {
        const int s = pi[e0 + e];
        const int d = pi[EE + e0 + e];
#pragma unroll
        for (int t = 0; t < 2; ++t) {
            const int f = lane + 32 * t;
            const float rI = rad[e * RDS + 3 * f + 0];
            const float rA = rad[e * RDS + 3 * f + 1];
            const float rS = rad[e * RDS + 3 * f + 2];
            const int g = d * FF + f;
            const int so = s * FF + f;
            atomicAdd(&M[0 * NF + so], rI * Q[0 * NF + g]);
#pragma unroll
            for (int c = 1; c < 4; ++c)
                atomicAdd(&M[c * NF + so], rA * Q[c * NF + g]);
#pragma unroll
            for (int c = 4; c < 10; ++c)
                atomicAdd(&M[c * NF + so], rS * Q[c * NF + g]);
        }
    }
}

// ---------------- K4: symmetric product + renorm + re-compact --------------
__global__ void combine_kernel(const float* __restrict__ M,
                               const float* __restrict__ Q,
                               const float* __restrict__ chg,
                               float* __restrict__ P) {
    int idx = blockIdx.x * blockDim.x + threadIdx.x;
    if (idx >= NF) return;
    const int n = idx >> 6;
    float mc[10], yc[10];
#pragma unroll
    for (int c = 0; c < 10; ++c) { mc[c] = M[c * NF + idx]; yc[c] = Q[c * NF + idx]; }
    float msg[9], Y[9], t1[9], t2[9];
    recon3(mc, msg);
    recon3(yc, Y);
    mm3(msg, Y, t1);
    mm3(Y, msg, t2);
    const float scale = 1.0f + 0.1f * chg[n];
    float T[9]; float tn = 0.f;
#pragma unroll
    for (int i = 0; i < 9; ++i) { T[i] = scale * (t1[i] + t2[i]); tn += T[i] * T[i]; }
    const float inv = 1.0f / (tn + 1.0f);
#pragma unroll
    for (int i = 0; i < 9; ++i) T[i] *= inv;
    float dm = (T[0] + T[4] + T[8]) * (1.0f / 3.0f);
    P[0 * NF + idx] = dm;
    P[1 * NF + idx] = 0.5f * (T[1] - T[3]);
    P[2 * NF + idx] = 0.5f * (T[2] - T[6]);
    P[3 * NF + idx] = 0.5f * (T[5] - T[7]);
    P[4 * NF + idx] = T[0] - dm;
    P[5 * NF + idx] = T[4] - dm;
    P[6 * NF + idx] = T[8] - dm;
    P[7 * NF + idx] = 0.5f * (T[1] + T[3]);
    P[8 * NF + idx] = 0.5f * (T[2] + T[6]);
    P[9 * NF + idx] = 0.5f * (T[5] + T[7]);
}

// ---------------- K6: out = Xn + dX + scale * dX@dX ------------------------
__global__ void finalize_kernel(const float* __restrict__ Q2,
                                const float* __restrict__ Xn,
                                const float* __restrict__ chg,
                                float* __restrict__ out) {
    int idx = blockIdx.x * blockDim.x + threadIdx.x;
    if (idx >= NF) return;
    const int n = idx >> 6;
    float dc[10];
#pragma unroll
    for (int c = 0; c < 10; ++c) dc[c] = Q2[c * NF + idx];
    float dX[9], prod[9];
    recon3(dc, dX);
    mm3(dX, dX, prod);
    const float scale = 1.0f + 0.1f * chg[n];
#pragma unroll
    for (int i = 0; i < 9; ++i)
        out[(size_t)idx * 9 + i] = Xn[(size_t)idx * 9 + i] + dX[i] + scale * prod[i];
}

// ---------------- launcher -------------------------------------------------
extern "C" void kernel_launch(void* const* d_in, const int* in_sizes, int n_in,
                              void* d_out, int out_size, void* d_ws, size_t ws_size,
                              hipStream_t stream) {
    const float* X   = (const float*)d_in[0];
    const int*   pi  = (const int*)  d_in[1];
    const float* dij = (const float*)d_in[2];
    const float* rbf = (const float*)d_in[3];
    const float* chg = (const float*)d_in[4];
    const float* W1  = (const float*)d_in[5];
    const float* b1  = (const float*)d_in[6];
    const float* W2  = (const float*)d_in[7];
    const float* b2  = (const float*)d_in[8];
    const float* W3  = (const float*)d_in[9];
    const float* b3  = (const float*)d_in[10];
    const float* L0  = (const float*)d_in[11];
    const float* L1  = (const float*)d_in[12];
    const float* L2  = (const float*)d_in[13];
    const float* L3  = (const float*)d_in[14];
    const float* L4  = (const float*)d_in[15];
    const float* L5  = (const float*)d_in[16];

    // workspace layout (floats): Xn[NF*9] | P[10*NF] | Q[10*NF] | M[10*NF]
    // total = 24,960,000 floats (~99.8 MB)
    float* ws = (float*)d_ws;
    float* Xn = ws;
    float* P  = Xn + (size_t)NF * 9;
    float* Q  = P + (size_t)10 * NF;
    float* M  = Q + (size_t)10 * NF;

    hipMemsetAsync(M, 0, (size_t)10 * NF * sizeof(float), stream);

    prep_kernel<<<(NF + 255) / 256, 256, 0, stream>>>(X, Xn, P);
    mix_kernel<<<dim3(NN / 16, 10), 128, 0, stream>>>(P, Q, L0, L1, L2);
    edge_kernel<<<EE / 16, 32, 0, stream>>>(rbf, dij, pi, W1, b1, W2, b2, W3, b3, Q, M);
    combine_kernel<<<(NF + 255) / 256, 256, 0, stream>>>(M, Q, chg, P);
    mix_kernel<<<dim3(NN / 16, 10), 128, 0, stream>>>(P, Q, L3, L4, L5);
    finalize_kernel<<<(NF + 255) / 256, 256, 0, stream>>>(Q, Xn, chg, (float*)d_out);
}